// CaT_3539053052503
// MI455X (gfx1250) — compile-verified
//
#include <hip/hip_runtime.h>

// Problem constants from the reference
#define NB 4096   // batch B
#define NN 64     // nodes N
#define NH 8      // heads H
#define HSZ 16    // head size HS
#define NL 3      // layers L

typedef __attribute__((ext_vector_type(2))) float v2f;
typedef __attribute__((ext_vector_type(8))) float v8f;

__device__ __forceinline__ bool mbit(unsigned lo, unsigned hi, int m) {
    unsigned w = (m < 32) ? lo : hi;
    return (w >> (m & 31)) & 1u;
}

// One block per batch element b. 256 threads = 8 waves (wave32); wave w handles head h=w.
// The scalar-collapsed attention: wei[n,m] = c_h * x[n] * x[m]; softmax over DAG-masked m;
// s[n] = sum_m p[n,m] x[m]. num/den computed as E(64x64) @ [x | 1] via v_wmma_f32_16x16x4_f32.
__global__ __launch_bounds__(256) void cat_kernel(
    const float* __restrict__ X,   // (B,N)
    const int*   __restrict__ dag, // (N,N)
    const float* __restrict__ Wk,  // (L,H,HS)
    const float* __restrict__ Wq,  // (L,H,HS)
    const float* __restrict__ Wv,  // (L,H,HS)
    const float* __restrict__ Wp,  // (L,H*HS,1)
    const float* __restrict__ bp,  // (L,1)
    const float* __restrict__ W1,  // (L,1,4)
    const float* __restrict__ b1,  // (L,4)
    const float* __restrict__ W2,  // (L,4,1)
    const float* __restrict__ b2,  // (L,1)
    const float* __restrict__ Wlm, // (1,1)
    const float* __restrict__ blm, // (1,)
    float* __restrict__ out)       // (B,N)
{
    __shared__ float    xs[NN];        // current x[b, n]
    __shared__ float    xmax_s[NN];    // masked max of x over m, per row n
    __shared__ float    xmin_s[NN];    // masked min
    __shared__ float    acc[NN];       // sum_h s[h,n] * wvp[h]
    __shared__ unsigned mbits[NN][2];  // maskT[n][m] = dag[m,n] != 0, as 64-bit mask

    const int b    = blockIdx.x;
    const int tid  = threadIdx.x;
    const int lane = tid & 31;
    const int head = tid >> 5;        // wave id == head id
    const int half = lane >> 4;       // 0: lanes 0-15, 1: lanes 16-31
    const int l15  = lane & 15;

    // ---- init: load x row, build DAG-transpose bitmask ----
    if (tid < NN) {
        xs[tid]  = X[b * NN + tid];
        acc[tid] = 0.0f;
        unsigned lo = 0u, hi = 0u;
        for (int m = 0; m < 32; ++m)  if (dag[m * NN + tid] != 0)        lo |= (1u << m);
        for (int m = 0; m < 32; ++m)  if (dag[(m + 32) * NN + tid] != 0) hi |= (1u << m);
        mbits[tid][0] = lo;
        mbits[tid][1] = hi;
    }
    __syncthreads();

    for (int l = 0; l < NL; ++l) {
        // ---- per-row masked min/max of x (for softmax max-subtraction) ----
        if (tid < NN) {
            unsigned lo = mbits[tid][0], hi = mbits[tid][1];
            float mx = -INFINITY, mn = INFINITY;
            for (int m = 0; m < NN; ++m) {
                if (mbit(lo, hi, m)) {
                    float v = xs[m];
                    mx = fmaxf(mx, v);
                    mn = fminf(mn, v);
                }
            }
            if ((lo | hi) == 0u) { mx = 0.0f; mn = 0.0f; }  // invalid row
            xmax_s[tid] = mx;
            xmin_s[tid] = mn;
        }
        __syncthreads();

        // ---- per-head constants: c_h = HS^-0.5 * <Wq_h,Wk_h>, wvp_h = <Wv_h, Wp_h> ----
        const float* wq = Wq + (l * NH + head) * HSZ;
        const float* wk = Wk + (l * NH + head) * HSZ;
        const float* wv = Wv + (l * NH + head) * HSZ;
        const float* wp = Wp + l * (NH * HSZ) + head * HSZ;
        float c = 0.0f, wvp = 0.0f;
        #pragma unroll
        for (int d = 0; d < HSZ; ++d) {
            c   += wq[d] * wk[d];
            wvp += wv[d] * wp[d];
        }
        c *= 0.25f;  // HS^{-1/2} = 1/4

        // ---- attention via WMMA: D(16x16) += E(16x4) x B(4x16), B = [x | 1 | 0...] ----
        #pragma unroll
        for (int r = 0; r < 4; ++r) {
            const int   n  = r * 16 + l15;     // A-matrix row M = l15 (both lane halves)
            const float xn = xs[n];
            const float a  = c * xn;
            const float M  = (a >= 0.0f) ? a * xmax_s[n] : a * xmin_s[n];
            const unsigned mlo = mbits[n][0], mhi = mbits[n][1];

            v8f C = {0.f, 0.f, 0.f, 0.f, 0.f, 0.f, 0.f, 0.f};
            #pragma unroll
            for (int kk = 0; kk < 16; ++kk) {
                // This lane supplies K = kk*4 + 2*half + {0,1} for both A and B
                const int m0 = kk * 4 + half * 2;
                const int m1 = m0 + 1;
                const float xm0 = xs[m0];
                const float xm1 = xs[m1];
                // A: masked exp matrix rows
                float e0 = mbit(mlo, mhi, m0) ? __expf(a * xm0 - M) : 0.0f;
                float e1 = mbit(mlo, mhi, m1) ? __expf(a * xm1 - M) : 0.0f;
                v2f A = {e0, e1};
                // B: column 0 = x_m (numerator), column 1 = 1 (denominator), rest 0
                float b0 = (l15 == 0) ? xm0 : ((l15 == 1) ? 1.0f : 0.0f);
                float b1v = (l15 == 0) ? xm1 : ((l15 == 1) ? 1.0f : 0.0f);
                v2f Bv = {b0, b1v};
                C = __builtin_amdgcn_wmma_f32_16x16x4_f32(
                        false, A, false, Bv, (short)0, C, false, false);
            }

            // D layout: VGPR v holds row M=v (lanes 0-15) / M=v+8 (lanes 16-31), col N=lane&15.
            // num at N==0 (lanes 0,16), den at N==1 (lanes 1,17).
            #pragma unroll
            for (int v = 0; v < 8; ++v) {
                float num = C[v];
                float den = __shfl_down(C[v], 1);   // lane L reads lane L+1 (col 1)
                if (l15 == 0) {
                    int   row = r * 16 + v + half * 8;
                    float s   = (den > 0.0f) ? (num / den) : 0.0f;
                    atomicAdd(&acc[row], s * wvp);  // ds_add_f32
                }
            }
        }
        __syncthreads();

        // ---- residual + projection + tiny MLP (per node) ----
        if (tid < NN) {
            float xv = xs[tid] + acc[tid] + bp[l];
            float xo = xv;
            #pragma unroll
            for (int j = 0; j < 4; ++j) {
                float hj = xv * W1[l * 4 + j] + b1[l * 4 + j];
                hj = fmaxf(hj, 0.0f);
                xo += hj * W2[l * 4 + j];
            }
            xo += b2[l];
            xs[tid]  = xo;
            acc[tid] = 0.0f;   // reset for next layer
        }
        __syncthreads();
    }

    // ---- LM head ----
    if (tid < NN) {
        out[b * NN + tid] = xs[tid] * Wlm[0] + blm[0];
    }
}

extern "C" void kernel_launch(void* const* d_in, const int* in_sizes, int n_in,
                              void* d_out, int out_size, void* d_ws, size_t ws_size,
                              hipStream_t stream) {
    const float* X   = (const float*)d_in[0];
    const int*   dag = (const int*)  d_in[1];
    const float* Wk  = (const float*)d_in[2];
    const float* Wq  = (const float*)d_in[3];
    const float* Wv  = (const float*)d_in[4];
    const float* Wp  = (const float*)d_in[5];
    const float* bp  = (const float*)d_in[6];
    const float* W1  = (const float*)d_in[7];
    const float* b1  = (const float*)d_in[8];
    const float* W2  = (const float*)d_in[9];
    const float* b2  = (const float*)d_in[10];
    const float* Wlm = (const float*)d_in[11];
    const float* blm = (const float*)d_in[12];
    float* out = (float*)d_out;

    cat_kernel<<<NB, 256, 0, stream>>>(X, dag, Wk, Wq, Wv, Wp, bp,
                                       W1, b1, W2, b2, Wlm, blm, out);
}